// QCONV2d_1x1_35880156791521
// MI455X (gfx1250) — compile-verified
//
#include <hip/hip_runtime.h>

// Problem constants (from reference): x [32, 64, 224, 224] f32, weights [1,1] f32,
// out [32, 64, 224, 224] f32 with out[b,j,:,:] = cos(x[b,63,:,:] + sum(weights)).
#define B_DIM   32
#define C_IN    64
#define C_OUT   64
#define HW      50176                     // 224*224
#define CH_STRIDE_BYTES (HW * 4)          // 200704 B between channel planes
#define CENTER_J 22                       // center so all 64 offsets fit IOFFSET (s24)
#define THREADS 256
#define FLOATS_PER_BLOCK 1024             // THREADS * 4
#define BLOCKS_PER_BATCH (HW / FLOATS_PER_BLOCK)   // 49

typedef float f4 __attribute__((ext_vector_type(4)));
typedef int   i4 __attribute__((ext_vector_type(4)));
typedef __attribute__((address_space(1))) i4* gi4_ptr;   // global int4*
typedef __attribute__((address_space(3))) i4* si4_ptr;   // LDS int4*

__global__ __launch_bounds__(THREADS)
void qconv_cos_bcast_kernel(const float* __restrict__ x,
                            const float* __restrict__ w,
                            float* __restrict__ out) {
    __shared__ f4 lds_tile[THREADS];   // 4 KB per block

    const int tid = threadIdx.x;
    const int t   = blockIdx.x;        // spatial tile within plane (0..48)
    const int b   = blockIdx.y;        // batch (0..31)

    // Spatial offset (in floats) this thread covers within the HxW plane.
    const size_t sp = (size_t)t * FLOATS_PER_BLOCK + (size_t)tid * 4;

    const float wsum = w[0];  // weights shape (1,1): sum == single element

    // One-shot read of the last input channel; non-temporal (never re-read).
    const float* xin_ptr = x + (size_t)b * ((size_t)C_IN * HW)
                             + (size_t)(C_IN - 1) * HW + sp;
    f4 xin = __builtin_nontemporal_load((const f4*)xin_ptr);

    f4 c;
    c.x = cosf(xin.x + wsum);
    c.y = cosf(xin.y + wsum);
    c.z = cosf(xin.z + wsum);
    c.w = cosf(xin.w + wsum);

    // Stage the cos tile in LDS once; then broadcast it to 64 channel planes.
    lds_tile[tid] = c;
    __syncthreads();   // ds writes visible before the async engine reads LDS

    // Base pointer centered at channel CENTER_J: every store target is then a
    // compile-time byte offset in [-22*200704, +41*200704] = [-4.4MB, +8.2MB],
    // all inside the signed 24-bit instruction offset -> one vaddr, 64 IOFFSETs.
    char* gmid = (char*)(out + (size_t)b * ((size_t)C_OUT * HW)
                             + (size_t)CENTER_J * HW + sp);

#if __has_builtin(__builtin_amdgcn_global_store_async_from_lds_b128)
    // CDNA5 async LDS->global path: 64 b128 async stores per lane from the same
    // 16 LDS bytes. Tracked by ASYNCcnt; wave holds no store data in VGPRs.
    si4_ptr lsrc = (si4_ptr)&lds_tile[tid];
#pragma unroll
    for (int j = 0; j < C_OUT; ++j) {
        gi4_ptr gdst = (gi4_ptr)(gmid + ((long)j - CENTER_J) * (long)CH_STRIDE_BYTES);
        __builtin_amdgcn_global_store_async_from_lds_b128(gdst, lsrc,
                                                          /*offset=*/0,
                                                          /*cpol=*/0);
    }
#if __has_builtin(__builtin_amdgcn_s_wait_asynccnt)
    __builtin_amdgcn_s_wait_asynccnt(0);
#else
    asm volatile("s_wait_asynccnt 0" ::: "memory");
#endif
#else
    // Fallback: direct non-temporal b128 stores (streaming writes, skip cache).
#pragma unroll
    for (int j = 0; j < C_OUT; ++j) {
        __builtin_nontemporal_store(
            c, (f4*)(gmid + ((long)j - CENTER_J) * (long)CH_STRIDE_BYTES));
    }
#endif
}

extern "C" void kernel_launch(void* const* d_in, const int* in_sizes, int n_in,
                              void* d_out, int out_size, void* d_ws, size_t ws_size,
                              hipStream_t stream) {
    const float* x = (const float*)d_in[0];   // [32, 64, 224, 224] f32
    const float* w = (const float*)d_in[1];   // [1, 1] f32
    float* out = (float*)d_out;               // [32, 64, 224, 224] f32

    (void)in_sizes; (void)n_in; (void)out_size; (void)d_ws; (void)ws_size;

    dim3 grid(BLOCKS_PER_BATCH, B_DIM);       // 49 x 32 = 1568 blocks
    dim3 block(THREADS);                      // 256 threads = 8 wave32 waves
    qconv_cos_bcast_kernel<<<grid, block, 0, stream>>>(x, w, out);
}